// GNN_10213432230422
// MI455X (gfx1250) — compile-verified
//
#include <hip/hip_runtime.h>
#include <hip/hip_bf16.h>

typedef __attribute__((ext_vector_type(2))) float v2f;
typedef __attribute__((ext_vector_type(8))) float v8f;

#define D_FEAT 128
#define LDS_PAD 132

// ---------------- degree ----------------
__global__ void deg_init_kernel(float* deg, int n) {
    int i = blockIdx.x * blockDim.x + threadIdx.x;
    if (i < n) deg[i] = 1.0f;  // self-loop
}

__global__ void deg_acc_kernel(const int* __restrict__ dst, float* deg, int e) {
    int i = blockIdx.x * blockDim.x + threadIdx.x;
    if (i < e) atomicAdd(&deg[dst[i]], 1.0f);
}

__global__ void deg_rsqrt_kernel(float* deg, int n) {
    int i = blockIdx.x * blockDim.x + threadIdx.x;
    if (i < n) deg[i] = rsqrtf(deg[i]);
}

// ---------------- zero ----------------
__global__ void zero_kernel(float* p, size_t n) {
    size_t i = (size_t)blockIdx.x * blockDim.x + threadIdx.x;
    size_t stride = (size_t)gridDim.x * blockDim.x;
    for (; i < n; i += stride) p[i] = 0.0f;
}

__global__ void stats_init_kernel(float* stats) {
    int i = threadIdx.x;
    if (i < 132) stats[i] = 0.0f;  // [0]=Z, [1..128]=pooled, [129]=max-enc(u32 0)
}

// ---------------- WMMA GEMM: H[N,128] = X[N,128] @ W[128,128] ----------------
// fp32 WMMA 16x16x4; block = 256 thr (8 waves); block computes 16 rows x 128 cols.
__global__ __launch_bounds__(256) void gemm_wmma_kernel(const float* __restrict__ X,
                                                        const float* __restrict__ W,
                                                        float* __restrict__ H, int n) {
    __shared__ float Ws[D_FEAT][LDS_PAD];
    __shared__ float Xs[16][LDS_PAD];
    const int tid = threadIdx.x;
    const int row0 = blockIdx.x * 16;

    for (int i = tid; i < D_FEAT * D_FEAT; i += 256)
        Ws[i >> 7][i & 127] = W[i];
    for (int i = tid; i < 16 * D_FEAT; i += 256) {
        int r = i >> 7, c = i & 127;
        int gr = row0 + r;
        Xs[r][c] = (gr < n) ? X[(size_t)gr * D_FEAT + c] : 0.0f;
    }
    __syncthreads();

    const int wave = tid >> 5;          // 0..7 -> 16-col tile
    const int lane = tid & 31;
    const int m = lane & 15;            // A row / B col within tile
    const int khalf = (lane >> 4) << 1; // 0 or 2
    const int ncol = (wave << 4) + m;

    v8f c = {};
    for (int k = 0; k < D_FEAT; k += 4) {
        const int kl = k + khalf;
        v2f a, b;
        a.x = Xs[m][kl];
        a.y = Xs[m][kl + 1];
        b.x = Ws[kl][ncol];
        b.y = Ws[kl + 1][ncol];
        c = __builtin_amdgcn_wmma_f32_16x16x4_f32(false, a, false, b, (short)0, c,
                                                  false, false);
    }

    // C/D layout: VGPR r -> row r (lanes 0-15) / row r+8 (lanes 16-31), col = lane%16
    const int srow = (lane < 16) ? 0 : 8;
    const int scol = (wave << 4) + (lane & 15);
    for (int r = 0; r < 8; ++r) {
        int gr = row0 + srow + r;
        if (gr < n) H[(size_t)gr * D_FEAT + scol] = c[r];
    }
}

// ---------------- edge scatter: AGG[dst] += H[src] * dinv[src]*dinv[dst] ----------------
// one wave per edge; 32 lanes x float4 = 128 channels
__global__ __launch_bounds__(256) void edge_scatter_kernel(const float* __restrict__ H,
                                                           float* __restrict__ AGG,
                                                           const int* __restrict__ src,
                                                           const int* __restrict__ dst,
                                                           const float* __restrict__ dinv,
                                                           int e) {
    int gtid = blockIdx.x * blockDim.x + threadIdx.x;
    int wid = gtid >> 5;
    int lane = threadIdx.x & 31;
    if (wid >= e) return;
    int s = src[wid];
    int d = dst[wid];
    float norm = dinv[s] * dinv[d];
    float4 v = reinterpret_cast<const float4*>(H + (size_t)s * D_FEAT)[lane];
    float* ap = AGG + (size_t)d * D_FEAT + lane * 4;
    atomicAdd(ap + 0, v.x * norm);
    atomicAdd(ap + 1, v.y * norm);
    atomicAdd(ap + 2, v.z * norm);
    atomicAdd(ap + 3, v.w * norm);
}

// ---------------- epilogue: AGG = relu(AGG + H*dinv^2 + b) ----------------
__global__ void post_kernel(const float* __restrict__ H, float* __restrict__ AGG,
                            const float* __restrict__ dinv, const float* __restrict__ b,
                            int n) {
    size_t i = (size_t)blockIdx.x * blockDim.x + threadIdx.x;
    size_t total = (size_t)n * D_FEAT;
    if (i >= total) return;
    int node = (int)(i >> 7);
    int col = (int)(i & 127);
    float sn = dinv[node];
    float v = AGG[i] + H[i] * (sn * sn) + b[col];
    AGG[i] = fmaxf(v, 0.0f);
}

// ---------------- attention logits + global max ----------------
__device__ inline unsigned enc_f(float f) {
    unsigned u = __float_as_uint(f);
    return (u & 0x80000000u) ? ~u : (u | 0x80000000u);
}
__device__ inline float dec_f(unsigned u) {
    return (u & 0x80000000u) ? __uint_as_float(u & 0x7FFFFFFFu)
                             : __uint_as_float(~u);
}

// one wave per node: l[i] = dot(H2[i], Wa) + ba
__global__ __launch_bounds__(256) void logits_kernel(const float* __restrict__ H2,
                                                     const float* __restrict__ Wa,
                                                     const float* __restrict__ ba,
                                                     float* __restrict__ logits,
                                                     unsigned* __restrict__ maxenc,
                                                     int n) {
    int gtid = blockIdx.x * blockDim.x + threadIdx.x;
    int node = gtid >> 5;
    int lane = threadIdx.x & 31;
    if (node >= n) return;
    float4 h = reinterpret_cast<const float4*>(H2 + (size_t)node * D_FEAT)[lane];
    float4 w = reinterpret_cast<const float4*>(Wa)[lane];
    float p = h.x * w.x + h.y * w.y + h.z * w.z + h.w * w.w;
    p += __shfl_xor(p, 16, 32);
    p += __shfl_xor(p, 8, 32);
    p += __shfl_xor(p, 4, 32);
    p += __shfl_xor(p, 2, 32);
    p += __shfl_xor(p, 1, 32);
    if (lane == 0) {
        float l = p + ba[0];
        logits[node] = l;
        atomicMax(maxenc, enc_f(l));
    }
}

// ---------------- pooled[d] = sum_i exp(l_i - max)*H2[i][d]; Z = sum exp ----------------
#define POOL_CHUNK 512
__global__ __launch_bounds__(128) void pool_kernel(const float* __restrict__ H2,
                                                   const float* __restrict__ logits,
                                                   const unsigned* __restrict__ maxenc,
                                                   float* __restrict__ pooled,
                                                   float* __restrict__ Zsum, int n) {
    __shared__ float wv[POOL_CHUNK];
    __shared__ float pz[128];
    const float gmax = dec_f(*maxenc);
    const int base = blockIdx.x * POOL_CHUNK;
    const int tid = threadIdx.x;

    float zloc = 0.0f;
    for (int j = tid; j < POOL_CHUNK; j += 128) {
        int nd = base + j;
        float w = (nd < n) ? expf(logits[nd] - gmax) : 0.0f;
        wv[j] = w;
        zloc += w;
    }
    pz[tid] = zloc;
    __syncthreads();

    int lim = n - base;
    if (lim > POOL_CHUNK) lim = POOL_CHUNK;
    float s = 0.0f;
    for (int j = 0; j < lim; ++j)
        s += wv[j] * H2[(size_t)(base + j) * D_FEAT + tid];
    atomicAdd(&pooled[tid], s);

    if (tid == 0) {
        float z = 0.0f;
        for (int i = 0; i < 128; ++i) z += pz[i];
        atomicAdd(Zsum, z);
    }
}

// ---------------- out[8] = (pooled @ Wf)/Z + bf ----------------
__global__ void final_kernel(const float* __restrict__ pooled, const float* __restrict__ Z,
                             const float* __restrict__ Wf, const float* __restrict__ bf,
                             float* __restrict__ out) {
    int j = threadIdx.x;
    if (j < 8) {
        float s = 0.0f;
        for (int d = 0; d < D_FEAT; ++d) s += pooled[d] * Wf[d * 8 + j];
        out[j] = s / (*Z) + bf[j];
    }
}

extern "C" void kernel_launch(void* const* d_in, const int* in_sizes, int n_in,
                              void* d_out, int out_size, void* d_ws, size_t ws_size,
                              hipStream_t stream) {
    const float* x  = (const float*)d_in[0];
    const int* ei   = (const int*)d_in[1];
    const float* W1 = (const float*)d_in[2];
    const float* b1 = (const float*)d_in[3];
    const float* W2 = (const float*)d_in[4];
    const float* b2 = (const float*)d_in[5];
    const float* Wa = (const float*)d_in[6];
    const float* ba = (const float*)d_in[7];
    const float* Wf = (const float*)d_in[8];
    const float* bf = (const float*)d_in[9];
    float* out = (float*)d_out;

    const int N = in_sizes[0] / D_FEAT;
    const int E = in_sizes[1] / 2;
    const int* src = ei;
    const int* dst = ei + E;

    // workspace carve (floats)
    float* ws = (float*)d_ws;
    float* dinv   = ws;                       // N  (deg then rsqrt in place)
    float* bufH   = dinv + N;                 // N*128 : transformed feats
    float* bufA   = bufH + (size_t)N * D_FEAT;// N*128 : aggregated / layer out
    float* logits = bufA + (size_t)N * D_FEAT;// N
    float* stats  = logits + N;               // [0]=Z, [1..128]=pooled, [129]=maxenc
    float* Zsum = stats;
    float* pooled = stats + 1;
    unsigned* maxenc = (unsigned*)(stats + 129);

    const size_t feat_elems = (size_t)N * D_FEAT;
    const int TPB = 256;

    // degrees -> dinv
    deg_init_kernel<<<(N + TPB - 1) / TPB, TPB, 0, stream>>>(dinv, N);
    deg_acc_kernel<<<(E + TPB - 1) / TPB, TPB, 0, stream>>>(dst, dinv, E);
    deg_rsqrt_kernel<<<(N + TPB - 1) / TPB, TPB, 0, stream>>>(dinv, N);

    // ---- layer 1 ----
    gemm_wmma_kernel<<<(N + 15) / 16, TPB, 0, stream>>>(x, W1, bufH, N);
    zero_kernel<<<4096, TPB, 0, stream>>>(bufA, feat_elems);
    edge_scatter_kernel<<<(E * 32 + TPB - 1) / TPB, TPB, 0, stream>>>(bufH, bufA, src,
                                                                      dst, dinv, E);
    post_kernel<<<(int)((feat_elems + TPB - 1) / TPB), TPB, 0, stream>>>(bufH, bufA,
                                                                         dinv, b1, N);

    // ---- layer 2 (bufA holds h1) ----
    gemm_wmma_kernel<<<(N + 15) / 16, TPB, 0, stream>>>(bufA, W2, bufH, N);
    zero_kernel<<<4096, TPB, 0, stream>>>(bufA, feat_elems);
    edge_scatter_kernel<<<(E * 32 + TPB - 1) / TPB, TPB, 0, stream>>>(bufH, bufA, src,
                                                                      dst, dinv, E);
    post_kernel<<<(int)((feat_elems + TPB - 1) / TPB), TPB, 0, stream>>>(bufH, bufA,
                                                                         dinv, b2, N);

    // ---- attention pooling over nodes (bufA holds h2) ----
    stats_init_kernel<<<1, 256, 0, stream>>>(stats);
    logits_kernel<<<(N * 32 + TPB - 1) / TPB, TPB, 0, stream>>>(bufA, Wa, ba, logits,
                                                                maxenc, N);
    pool_kernel<<<(N + POOL_CHUNK - 1) / POOL_CHUNK, 128, 0, stream>>>(bufA, logits,
                                                                       maxenc, pooled,
                                                                       Zsum, N);
    final_kernel<<<1, 32, 0, stream>>>(pooled, Zsum, Wf, bf, out);
}